// Coords2Eps_88871463289418
// MI455X (gfx1250) — compile-verified
//
#include <hip/hip_runtime.h>
#include <math.h>

#define BOX 128
#define NATOMS 2048
#define NBATCH 8
#define EPS_IN 6.5f
#define EPS_OUT 79.0f

typedef float v2f __attribute__((ext_vector_type(2)));
typedef float v8f __attribute__((ext_vector_type(8)));

// ---------------------------------------------------------------------------
// Kernel 1: zero the rho accumulator (d_out is poisoned by the harness).
// ---------------------------------------------------------------------------
__global__ void Coords2Eps_zero_kernel(float4* __restrict__ out, int n4) {
    int i = blockIdx.x * blockDim.x + threadIdx.x;
    if (i < n4) out[i] = make_float4(0.f, 0.f, 0.f, 0.f);
}

// ---------------------------------------------------------------------------
// Kernel 2: separable Gaussian scatter.
// Each wave32 handles TWO atoms. The 8x8 gx (x) gy outer products of both
// atoms are computed by a single V_WMMA_F32_16X16X4_F32 (rank-1 update:
// A col0 = [gx_a0 | gx_a1], B row0 = [gy_a0 | gy_a1]; diagonal 8x8 blocks of
// D are the per-atom outer products). Scatter phase uses all 32 lanes:
// lane -> (atom = lane>>4, j = lane&7, k-half = (lane>>3)&1), value pulled
// from D with __shfl, times 4 private gz factors, atomically added into the
// L2-resident rho grid.
// ---------------------------------------------------------------------------
__global__ __launch_bounds__(256) void Coords2Eps_scatter_kernel(
    const float* __restrict__ coords,
    const float* __restrict__ params,
    const int*   __restrict__ num_atoms,
    float*       __restrict__ rho) {

    const int lane = threadIdx.x & 31;
    const int gw   = (blockIdx.x * 256 + threadIdx.x) >> 5;  // global wave id == pair id
    const int b    = gw >> 10;                               // 1024 pairs per batch (N=2048)
    const int a0   = (gw & 1023) * 2;
    const int a1   = a0 + 1;

    const float* cb = coords + (size_t)b * 3 * NATOMS;
    const float* pb = params + (size_t)b * 2 * NATOMS;
    const int    na = num_atoms[b];

    const float ax0 = cb[3 * a0 + 0], ay0 = cb[3 * a0 + 1], az0 = cb[3 * a0 + 2];
    const float ax1 = cb[3 * a1 + 0], ay1 = cb[3 * a1 + 1], az1 = cb[3 * a1 + 2];
    const float sg0 = pb[2 * a0 + 1], sg1 = pb[2 * a1 + 1];
    const float iv0 = 0.5f / (sg0 * sg0), iv1 = 0.5f / (sg1 * sg1);
    const float m0  = (a0 < na) ? 1.f : 0.f;
    const float m1  = (a1 < na) ? 1.f : 0.f;

    const int bx0 = (int)floorf(ax0) - 3, by0 = (int)floorf(ay0) - 3, bz0 = (int)floorf(az0) - 3;
    const int bx1 = (int)floorf(ax1) - 3, by1 = (int)floorf(ay1) - 3, bz1 = (int)floorf(az1) - 3;

    // Build A (16x4, only K=0 column nonzero) and B (4x16, only K=0 row
    // nonzero). 32-bit A layout: VGPR0 lanes 0-15 hold K=0 for M=lane,
    // lanes 16-31 hold K=2 (zero here). B analogous.
    float ga = 0.f, gb = 0.f;
    if (lane < 16) {
        const int   sel = lane >> 3, off = lane & 7;
        const float ax = sel ? ax1 : ax0, ay = sel ? ay1 : ay0;
        const float iv = sel ? iv1 : iv0, mm = sel ? m1 : m0;
        const int   bx = sel ? bx1 : bx0, by = sel ? by1 : by0;
        const int   xi = bx + off, yj = by + off;
        const float dx = (float)xi - ax, dy = (float)yj - ay;
        ga = (xi >= 0 && xi < BOX) ? mm * __expf(-dx * dx * iv) : 0.f;
        gb = (yj >= 0 && yj < BOX) ?      __expf(-dy * dy * iv) : 0.f;
    }
    v2f A  = {ga, 0.f};
    v2f Bm = {gb, 0.f};
    v8f C  = {};
    // D[i][j] = gx_pair[i] * gy_pair[j]  (diagonal 8x8 blocks are the atoms)
    v8f D = __builtin_amdgcn_wmma_f32_16x16x4_f32(
        /*neg_a=*/false, A, /*neg_b=*/false, Bm,
        /*c_mod=*/(short)0, C, /*reuse_a=*/false, /*reuse_b=*/false);

    // Per-lane gz factors for the scatter phase.
    const int   as    = lane >> 4;             // which atom this lane scatters
    const int   kbase = ((lane >> 3) & 1) * 4; // which half of the k-window
    const float azs = as ? az1 : az0;
    const float ivs = as ? iv1 : iv0;
    const int   bxs = as ? bx1 : bx0, bys = as ? by1 : by0, bzs = as ? bz1 : bz0;

    float gz[4];
#pragma unroll
    for (int kk = 0; kk < 4; ++kk) {
        const int   zz = bzs + kbase + kk;
        const float dz = (float)zz - azs;
        gz[kk] = (zz >= 0 && zz < BOX) ? __expf(-dz * dz * ivs) : 0.f;
    }

    const int j   = lane & 7;
    const int src = (lane < 16) ? j : 24 + j;  // D source lane for (atom, j)
    float* rb = rho + (size_t)b * BOX * BOX * BOX;

#pragma unroll
    for (int i = 0; i < 8; ++i) {
        const float gxy = __shfl(D[i], src, 32);
        if (gxy != 0.f) {
            const int xi = bxs + i, yj = bys + j;
            float* base = rb + ((size_t)(xi * BOX + yj) * BOX + (bzs + kbase));
#pragma unroll
            for (int kk = 0; kk < 4; ++kk) {
                if (gz[kk] != 0.f) atomicAdd(base + kk, gxy * gz[kk]);
            }
        }
    }
}

// ---------------------------------------------------------------------------
// Kernel 3: eps = exp(-rho) * (EPS_OUT - EPS_IN) + EPS_IN, in place, float4.
// ---------------------------------------------------------------------------
__global__ void Coords2Eps_finalize_kernel(float4* __restrict__ out, int n4) {
    int i = blockIdx.x * blockDim.x + threadIdx.x;
    if (i < n4) {
        float4 v = out[i];
        v.x = __expf(-v.x) * (EPS_OUT - EPS_IN) + EPS_IN;
        v.y = __expf(-v.y) * (EPS_OUT - EPS_IN) + EPS_IN;
        v.z = __expf(-v.z) * (EPS_OUT - EPS_IN) + EPS_IN;
        v.w = __expf(-v.w) * (EPS_OUT - EPS_IN) + EPS_IN;
        out[i] = v;
    }
}

extern "C" void kernel_launch(void* const* d_in, const int* in_sizes, int n_in,
                              void* d_out, int out_size, void* d_ws, size_t ws_size,
                              hipStream_t stream) {
    const float* coords    = (const float*)d_in[0];  // (B, 3N) f32
    const float* params    = (const float*)d_in[1];  // (B, N, 2) f32
    const int*   num_atoms = (const int*)d_in[2];    // (B,) i32
    float*       out       = (float*)d_out;          // (B, 128^3) f32

    const int total = NBATCH * BOX * BOX * BOX;      // 16,777,216
    const int n4    = total / 4;                     // 4,194,304

    Coords2Eps_zero_kernel<<<n4 / 256, 256, 0, stream>>>((float4*)out, n4);

    const int pairs  = NBATCH * NATOMS / 2;          // 8192 wave-tasks
    const int blocks = pairs / 8;                    // 8 waves per 256-thread block
    Coords2Eps_scatter_kernel<<<blocks, 256, 0, stream>>>(coords, params, num_atoms, out);

    Coords2Eps_finalize_kernel<<<n4 / 256, 256, 0, stream>>>((float4*)out, n4);
}